// BipartiteGNN_61710090109663
// MI455X (gfx1250) — compile-verified
//
#include <hip/hip_runtime.h>

typedef __attribute__((ext_vector_type(2))) float v2f;
typedef __attribute__((ext_vector_type(8))) float v8f;

#define N_NODES 50000
#define N_EDGES 400000
#define N_GRAPH 128
#define HID     256
#define FIN     4
#define GFD     16
#define AOUT    64

// ---------------------------------------------------------------------------
// Generic fp32 WMMA GEMM:  Y[M,Nc] = A[M,K] @ W[K,Nc] (+bias)(+relu)
// One wave -> one 16x64 output tile (4 accumulators of v8f), software-
// pipelined so the k+4 fragments are in flight while the k-step WMMAs issue.
// Optional fused second output: Y2[r,c] = dinv[r]^2 * Y[r,c]  (GCN self-loop).
// Uses V_WMMA_F32_16X16X4_F32 (full fp32, matches reference precision).
// Fragment layouts per CDNA5 ISA 7.12.2 (half = lane>>4, m/n = lane&15):
//   A 16x4 : VGPR0/1 = A[m][2*half], A[m][2*half+1]
//   B 4x16 : VGPR0/1 = B[2*half][n], B[2*half+1][n]
//   C 16x16: VGPR r -> row r (lanes 0-15) / row r+8 (lanes 16-31), col n
// Requires M%16==0, K%4==0, Nc%64==0 (true for every call site here).
// ---------------------------------------------------------------------------
__global__ void gemm_wmma_f32(const float* __restrict__ A,
                              const float* __restrict__ W,
                              const float* __restrict__ bias,
                              float* __restrict__ Y,
                              float* __restrict__ Y2,
                              const float* __restrict__ dinv,
                              int M, int K, int Nc, int fuseRelu)
{
    const int wave   = threadIdx.x >> 5;
    const int lane   = threadIdx.x & 31;
    const int tilesN = Nc >> 6;
    const int tiles  = (M >> 4) * tilesN;
    const int tid    = blockIdx.x * 8 + wave;
    if (tid >= tiles) return;                 // wave-uniform: EXEC stays all-1s
    const int tm   = tid / tilesN;
    const int tn   = tid % tilesN;
    const int m    = lane & 15;
    const int half = lane >> 4;
    const int cb   = tn << 6;

    v8f acc[4] = {v8f{}, v8f{}, v8f{}, v8f{}};

    const float* arow  = A + (size_t)((tm << 4) + m) * K + 2 * half;
    const float* wbase = W + (size_t)(2 * half) * Nc + cb + m;

    // ---- prologue: fragments for k = 0 ----
    v2f a = *(const v2f*)(arow);
    v2f b[4];
#pragma unroll
    for (int j = 0; j < 4; ++j) {
        b[j].x = wbase[j * 16];
        b[j].y = wbase[Nc + j * 16];
    }

    // ---- steady state: prefetch k, compute k-4 ----
    for (int k = 4; k < K; k += 4) {
        v2f an = *(const v2f*)(arow + k);
        const float* w0 = wbase + (size_t)k * Nc;
        v2f bn[4];
#pragma unroll
        for (int j = 0; j < 4; ++j) {
            bn[j].x = w0[j * 16];
            bn[j].y = w0[Nc + j * 16];
        }
#pragma unroll
        for (int j = 0; j < 4; ++j)
            acc[j] = __builtin_amdgcn_wmma_f32_16x16x4_f32(
                false, a, false, b[j], (short)0, acc[j], false, false);
        a = an;
#pragma unroll
        for (int j = 0; j < 4; ++j) b[j] = bn[j];
    }

    // ---- epilogue: last k-step ----
#pragma unroll
    for (int j = 0; j < 4; ++j)
        acc[j] = __builtin_amdgcn_wmma_f32_16x16x4_f32(
            false, a, false, b[j], (short)0, acc[j], false, false);

    // ---- store (optionally fused bias/relu and self-loop init) ----
    float dsq[8];
    if (Y2) {
#pragma unroll
        for (int r = 0; r < 8; ++r) {
            float di = dinv[(tm << 4) + r + 8 * half];
            dsq[r] = di * di;
        }
    }
#pragma unroll
    for (int j = 0; j < 4; ++j) {
        const int c  = cb + j * 16 + m;
        const float bb = bias ? bias[c] : 0.0f;
#pragma unroll
        for (int r = 0; r < 8; ++r) {
            const int orow = (tm << 4) + r + 8 * half;
            float v = acc[j][r] + bb;
            if (fuseRelu) v = fmaxf(v, 0.0f);
            Y[(size_t)orow * Nc + c] = v;
            if (Y2) Y2[(size_t)orow * Nc + c] = dsq[r] * v;
        }
    }
}

// ---------------------------------------------------------------------------
// Degree / normalization kernels
// ---------------------------------------------------------------------------
__global__ void deg_init_kernel(float* __restrict__ deg)
{
    int i = blockIdx.x * blockDim.x + threadIdx.x;
    if (i < N_NODES) deg[i] = 1.0f;                 // self-loop weight
}

__global__ void deg_edge_kernel(const int* __restrict__ dst,
                                const float* __restrict__ ew,
                                float* __restrict__ deg)
{
    int e = blockIdx.x * blockDim.x + threadIdx.x;
    if (e < N_EDGES) atomicAdd(&deg[dst[e]], ew[e]);
}

__global__ void deg_finalize_kernel(float* __restrict__ deg)
{
    int i = blockIdx.x * blockDim.x + threadIdx.x;
    if (i < N_NODES) {
        float d = deg[i];
        deg[i] = (d > 0.0f) ? rsqrtf(d) : 0.0f;
    }
}

// ---------------------------------------------------------------------------
// Edge scatter: agg[dst] += dinv[src]*ew*dinv[dst] * y[src]   (64 lanes/edge)
// ---------------------------------------------------------------------------
__global__ void scatter_kernel(const float* __restrict__ y,
                               const int* __restrict__ src,
                               const int* __restrict__ dst,
                               const float* __restrict__ ew,
                               const float* __restrict__ dinv,
                               float* __restrict__ agg)
{
    int idx = blockIdx.x * blockDim.x + threadIdx.x;
    int e  = idx >> 6;
    if (e >= N_EDGES) return;
    int c4 = (idx & 63) << 2;
    int s = src[e], d = dst[e];
    float coef = dinv[s] * ew[e] * dinv[d];
    const float4 v = *(const float4*)(y + (size_t)s * HID + c4);
    float* o = agg + (size_t)d * HID + c4;
    atomicAdd(o + 0, coef * v.x);
    atomicAdd(o + 1, coef * v.y);
    atomicAdd(o + 2, coef * v.z);
    atomicAdd(o + 3, coef * v.w);
}

// ---------------------------------------------------------------------------
// h[n][c] = relu(agg[n][c] + b[c])
// ---------------------------------------------------------------------------
__global__ void bias_relu_kernel(const float* __restrict__ agg,
                                 const float* __restrict__ b,
                                 float* __restrict__ h)
{
    int idx = blockIdx.x * blockDim.x + threadIdx.x;
    int n  = idx >> 6;
    int c4 = (idx & 63) << 2;
    if (n >= N_NODES) return;
    const float4 v = *(const float4*)(agg + (size_t)n * HID + c4);
    const float4 bb = *(const float4*)(b + c4);
    float4 o;
    o.x = fmaxf(v.x + bb.x, 0.0f);
    o.y = fmaxf(v.y + bb.y, 0.0f);
    o.z = fmaxf(v.z + bb.z, 0.0f);
    o.w = fmaxf(v.w + bb.w, 0.0f);
    *(float4*)(h + (size_t)n * HID + c4) = o;
}

__global__ void zero_kernel(float* __restrict__ p, int count)
{
    int i = blockIdx.x * blockDim.x + threadIdx.x;
    if (i < count) p[i] = 0.0f;
}

// ---------------------------------------------------------------------------
// Per-graph mean pool (atomic accumulation)
// ---------------------------------------------------------------------------
__global__ void pool_kernel(const float* __restrict__ h,
                            const int* __restrict__ batch,
                            float* __restrict__ gsum,
                            float* __restrict__ gcnt)
{
    int idx = blockIdx.x * blockDim.x + threadIdx.x;
    int n  = idx >> 6;
    int c4 = (idx & 63) << 2;
    if (n >= N_NODES) return;
    int g = batch[n];
    const float4 v = *(const float4*)(h + (size_t)n * HID + c4);
    float* o = gsum + (size_t)g * HID + c4;
    atomicAdd(o + 0, v.x);
    atomicAdd(o + 1, v.y);
    atomicAdd(o + 2, v.z);
    atomicAdd(o + 3, v.w);
    if (c4 == 0) atomicAdd(&gcnt[g], 1.0f);
}

// combined[g] = [ gsum/cnt , gemb ]
__global__ void combine_kernel(const float* __restrict__ gsum,
                               const float* __restrict__ gcnt,
                               const float* __restrict__ gemb,
                               float* __restrict__ comb)
{
    int idx = blockIdx.x * blockDim.x + threadIdx.x;
    if (idx >= N_GRAPH * 2 * HID) return;
    int g = idx / (2 * HID);
    int c = idx % (2 * HID);
    float v;
    if (c < HID) {
        float cnt = fmaxf(gcnt[g], 1.0f);
        v = gsum[(size_t)g * HID + c] / cnt;
    } else {
        v = gemb[(size_t)g * HID + (c - HID)];
    }
    comb[idx] = v;
}

// ---------------------------------------------------------------------------
extern "C" void kernel_launch(void* const* d_in, const int* in_sizes, int n_in,
                              void* d_out, int out_size, void* d_ws, size_t ws_size,
                              hipStream_t stream)
{
    (void)in_sizes; (void)n_in; (void)out_size; (void)ws_size;
    const float* x     = (const float*)d_in[0];
    const int*   eidx  = (const int*)d_in[1];
    const int*   batch = (const int*)d_in[2];
    const float* gf    = (const float*)d_in[3];
    const float* ew    = (const float*)d_in[4];
    const float* W_emb = (const float*)d_in[5];  const float* b_emb = (const float*)d_in[6];
    const float* W1    = (const float*)d_in[7];  const float* b1    = (const float*)d_in[8];
    const float* W2    = (const float*)d_in[9];  const float* b2    = (const float*)d_in[10];
    const float* Wg    = (const float*)d_in[11]; const float* bg    = (const float*)d_in[12];
    const float* Wh1   = (const float*)d_in[13]; const float* bh1   = (const float*)d_in[14];
    const float* Wh2   = (const float*)d_in[15]; const float* bh2   = (const float*)d_in[16];
    const int* src = eidx;
    const int* dst = eidx + N_EDGES;

    char* p = (char*)d_ws;
    float* h    = (float*)p; p += (size_t)N_NODES * HID * sizeof(float);
    float* y    = (float*)p; p += (size_t)N_NODES * HID * sizeof(float);
    float* agg  = (float*)p; p += (size_t)N_NODES * HID * sizeof(float);
    float* dinv = (float*)p; p += (size_t)N_NODES * sizeof(float);
    float* gsum = (float*)p; p += (size_t)N_GRAPH * HID * sizeof(float);
    float* gcnt = (float*)p; p += (size_t)N_GRAPH * sizeof(float);
    float* gemb = (float*)p; p += (size_t)N_GRAPH * HID * sizeof(float);
    float* comb = (float*)p; p += (size_t)N_GRAPH * 2 * HID * sizeof(float);
    float* t1   = (float*)p; p += (size_t)N_GRAPH * HID * sizeof(float);

    auto gemm = [&](const float* A, const float* W, const float* bias, float* Y,
                    float* Y2, const float* dv, int M, int K, int Nc, int relu) {
        int tiles  = (M / 16) * (Nc / 64);
        int blocks = (tiles + 7) / 8;
        gemm_wmma_f32<<<blocks, 256, 0, stream>>>(A, W, bias, Y, Y2, dv, M, K, Nc, relu);
    };

    // 1) node embedding: h = relu(x @ W_emb + b_emb)
    gemm(x, W_emb, b_emb, h, nullptr, nullptr, N_NODES, FIN, HID, 1);

    // 2) symmetric normalization factors
    deg_init_kernel<<<(N_NODES + 255) / 256, 256, 0, stream>>>(dinv);
    deg_edge_kernel<<<(N_EDGES + 255) / 256, 256, 0, stream>>>(dst, ew, dinv);
    deg_finalize_kernel<<<(N_NODES + 255) / 256, 256, 0, stream>>>(dinv);

    // 3) two GCN layers: GEMM (+fused self-loop init of agg) -> scatter -> bias+relu
    const int nodeC4 = N_NODES * (HID / 4);
    const int edgeC4 = N_EDGES * (HID / 4);
    for (int l = 0; l < 2; ++l) {
        const float* W = l ? W2 : W1;
        const float* b = l ? b2 : b1;
        gemm(h, W, nullptr, y, agg, dinv, N_NODES, HID, HID, 0);
        scatter_kernel<<<(edgeC4 + 255) / 256, 256, 0, stream>>>(y, src, dst, ew, dinv, agg);
        bias_relu_kernel<<<(nodeC4 + 255) / 256, 256, 0, stream>>>(agg, b, h);
    }

    // 4) global mean pool (gsum & gcnt are contiguous -> one zero pass)
    zero_kernel<<<(N_GRAPH * HID + N_GRAPH + 255) / 256, 256, 0, stream>>>(gsum, N_GRAPH * HID + N_GRAPH);
    pool_kernel<<<(nodeC4 + 255) / 256, 256, 0, stream>>>(h, batch, gsum, gcnt);

    // 5) global feature embedding + concat
    gemm(gf, Wg, bg, gemb, nullptr, nullptr, N_GRAPH, GFD, HID, 1);
    combine_kernel<<<(N_GRAPH * 2 * HID + 255) / 256, 256, 0, stream>>>(gsum, gcnt, gemb, comb);

    // 6) head MLP -> d_out [128, 64]
    gemm(comb, Wh1, bh1, t1, nullptr, nullptr, N_GRAPH, 2 * HID, HID, 1);
    gemm(t1, Wh2, bh2, (float*)d_out, nullptr, nullptr, N_GRAPH, HID, AOUT, 0);
}